// LBPModule_19902878450061
// MI455X (gfx1250) — compile-verified
//
#include <hip/hip_runtime.h>

typedef __attribute__((ext_vector_type(2))) float v2f;
typedef __attribute__((ext_vector_type(8))) float v8f;

// ---------------- geometry ----------------
constexpr int B_   = 16;
constexpr int HP   = 257, WP = 257;      // pooled / LBP spatial size
constexpr int NPIX = HP * WP;            // 66049
constexpr int NB4  = 64;                 // B*4 groups
constexpr int NPLANE = 128;              // B*8 (b4, ch) planes for min/max
constexpr int NSM  = 3584;               // 64*56 small-grid elements per (b,p)
constexpr int NPOS = 16 * NSM;           // 57344 positions for channel projection
constexpr int NTILE = NPOS / 16;         // 3584 WMMA tiles (224 per batch)

// ---------------- workspace layout (float offsets) ----------------
constexpr size_t OFF_S      = 0;                          // 16*257*257 = 1056784
constexpr size_t OFF_MN     = 1056800;                    // 128 uint (flip-encoded)
constexpr size_t OFF_MX     = OFF_MN + 128;               // 128 uint
constexpr size_t OFF_MEAS   = OFF_MX + 128;               // 64*9*2 = 1152
constexpr size_t OFF_HIST   = OFF_MEAS + 1152;            // 64*64 = 4096
constexpr size_t OFF_HSUM   = OFF_HIST + 4096;            // 1 (padded 32)
constexpr size_t OFF_STATS  = OFF_HSUM + 32;              // 64*2 (mean, inv_std)
constexpr size_t OFF_HSMALL = OFF_STATS + 128;            // 16*4*64*56 = 229376
constexpr size_t OFF_H2     = OFF_HSMALL + 229376;        // 57344*16 = 917504

// LBP neighbor offsets (kernel o -> (dy,dx) relative to center in S space)
__device__ __constant__ int LBP_DY[8] = {-1, 1, -1, 1, -1, 1, 0, 0};
__device__ __constant__ int LBP_DX[8] = {-1, 1,  0, 0,  1,-1, 1,-1};

#define DEVINL __device__ __forceinline__

DEVINL float leaky01(float v) { return (v >= 0.f) ? v : 0.01f * v; }

// order-preserving float<->uint for atomic min/max
DEVINL unsigned encf(float f) {
  unsigned u = __float_as_uint(f);
  return (u & 0x80000000u) ? ~u : (u | 0x80000000u);
}
DEVINL float decf(unsigned e) {
  return (e & 0x80000000u) ? __uint_as_float(e & 0x7FFFFFFFu)
                           : __uint_as_float(~e);
}

DEVINL float loadS(const float* S, int b, int y, int x) {
  if ((unsigned)y < (unsigned)HP && (unsigned)x < (unsigned)WP)
    return S[(b * HP + y) * WP + x];
  return 0.f;
}

// ---------------- K_init: reset accumulators ----------------
__global__ __launch_bounds__(256) void k_init(float* ws) {
  unsigned* mnb = (unsigned*)(ws + OFF_MN);
  unsigned* mxb = (unsigned*)(ws + OFF_MX);
  float* hist = ws + OFF_HIST;
  int t = threadIdx.x;
  for (int i = t; i < 4096; i += 256) hist[i] = 0.f;
  if (t < NPLANE) { mnb[t] = 0xFFFFFFFFu; mxb[t] = 0u; }
  if (t == 0) ws[OFF_HSUM] = 0.f;
}

// ---------------- K0: S = sum_c avgpool2x2(x), shape (16,257,257) ----------------
__global__ __launch_bounds__(256) void k_S(const float* __restrict__ x, float* __restrict__ ws) {
  int rem = blockIdx.x * 256 + threadIdx.x;
  if (rem >= NPIX) return;
  int b = blockIdx.y;
  int y = rem / WP, xx = rem % WP;
  float acc = 0.f;
  int r0 = y - 1, c0 = xx - 1;
  #pragma unroll
  for (int c = 0; c < 3; ++c) {
    const float* base = x + ((size_t)(b * 3 + c)) * 256 * 256;
    #pragma unroll
    for (int dr = 0; dr < 2; ++dr) {
      int r = r0 + dr;
      if ((unsigned)r >= 256u) continue;
      #pragma unroll
      for (int dc = 0; dc < 2; ++dc) {
        int cc = c0 + dc;
        if ((unsigned)cc >= 256u) continue;
        acc += base[r * 256 + cc];
      }
    }
  }
  ws[OFF_S + (size_t)(b * HP + y) * WP + xx] = acc * 0.25f;
}

// ---------------- K1: per-(b4,ch) min/max of clipped xl ----------------
__global__ __launch_bounds__(256) void k_minmax(float* ws, const float* clo, const float* chi) {
  const float* S = ws + OFF_S;
  unsigned* mnb = (unsigned*)(ws + OFF_MN);
  unsigned* mxb = (unsigned*)(ws + OFF_MX);
  int plane = blockIdx.y;                // b*8 + o
  int b = plane >> 3, o = plane & 7;
  int dy = LBP_DY[o], dx = LBP_DX[o];
  float cl = clo[0], ch = chi[0];
  float vmin = 3.0e38f, vmax = -3.0e38f;
  for (int idx = blockIdx.x * 256 + threadIdx.x; idx < NPIX; idx += 32 * 256) {
    int h = idx / WP, w = idx % WP;
    float v = S[(b * HP + h) * WP + w] - loadS(S, b, h + dy, w + dx);
    v = fminf(fmaxf(v, cl), ch);
    vmin = fminf(vmin, v);
    vmax = fmaxf(vmax, v);
  }
  __shared__ float smn[256], smx[256];
  int t = threadIdx.x;
  smn[t] = vmin; smx[t] = vmax;
  __syncthreads();
  for (int s = 128; s > 0; s >>= 1) {
    if (t < s) { smn[t] = fminf(smn[t], smn[t + s]); smx[t] = fmaxf(smx[t], smx[t + s]); }
    __syncthreads();
  }
  if (t == 0) {
    atomicMin(&mnb[plane], encf(smn[0]));
    atomicMax(&mxb[plane], encf(smx[0]));
  }
}

// ---------------- K2: measure[b4][q][ch] = q*cd + mn ----------------
__global__ __launch_bounds__(128) void k_measure(float* ws) {
  int t = threadIdx.x;               // 0..127 = b4*2 + ch
  unsigned* mnb = (unsigned*)(ws + OFF_MN);
  unsigned* mxb = (unsigned*)(ws + OFF_MX);
  float mn = decf(mnb[t]), mx = decf(mxb[t]);
  float cd = (mx - mn) / 8.0f;
  float* meas = ws + OFF_MEAS;
  int b4 = t >> 1, ch = t & 1;
  #pragma unroll
  for (int q = 0; q <= 8; ++q)
    meas[b4 * 18 + q * 2 + ch] = (float)q * cd + mn;
}

// ---------------- K3: weighted 2D histogram per b4 ----------------
__global__ __launch_bounds__(256) void k_hist(float* ws, const float* clo, const float* chi) {
  const float* S = ws + OFF_S;
  const float* measG = ws + OFF_MEAS;
  float* hist = ws + OFF_HIST;
  int b4 = blockIdx.y;
  int b = b4 >> 2, pair = b4 & 3;
  int o0 = pair * 2, o1 = o0 + 1;
  int dy0 = LBP_DY[o0], dx0 = LBP_DX[o0];
  int dy1 = LBP_DY[o1], dx1 = LBP_DX[o1];
  float cl = clo[0], ch = chi[0];
  __shared__ float meas[18];
  __shared__ float lh[64];
  int t = threadIdx.x;
  if (t < 18) meas[t] = measG[b4 * 18 + t];
  if (t < 64) lh[t] = 0.f;
  __syncthreads();
  for (int idx = blockIdx.x * 256 + t; idx < NPIX; idx += 32 * 256) {
    int h = idx / WP, w = idx % WP;
    float Sc = S[(b * HP + h) * WP + w];
    float v0 = fminf(fmaxf(Sc - loadS(S, b, h + dy0, w + dx0), cl), ch);
    float v1 = fminf(fmaxf(Sc - loadS(S, b, h + dy1, w + dx1), cl), ch);
    unsigned m0 = 0, m1 = 0;
    #pragma unroll
    for (int i = 0; i < 8; ++i) {
      m0 |= (v0 >= meas[i * 2 + 0] && v0 <= meas[i * 2 + 2]) ? (1u << i) : 0u;
      m1 |= (v1 >= meas[i * 2 + 1] && v1 <= meas[i * 2 + 3]) ? (1u << i) : 0u;
    }
    float prod = v0 * v1;
    for (unsigned a = m0; a; a &= a - 1) {
      int i = __ffs(a) - 1;
      for (unsigned bb = m1; bb; bb &= bb - 1) {
        int j = __ffs(bb) - 1;
        atomicAdd(&lh[i * 8 + j], prod);
      }
    }
  }
  __syncthreads();
  if (t < 64) {
    float f = lh[t];
    if (f != 0.f) atomicAdd(&hist[b4 * 64 + t], f);
  }
}

// ---------------- K4: total sum of hist ----------------
__global__ __launch_bounds__(256) void k_histsum(float* ws) {
  const float* hist = ws + OFF_HIST;
  __shared__ float sh[256];
  int t = threadIdx.x;
  float s = 0.f;
  for (int i = t; i < 4096; i += 256) s += hist[i];
  sh[t] = s;
  __syncthreads();
  for (int k = 128; k > 0; k >>= 1) {
    if (t < k) sh[t] += sh[t + k];
    __syncthreads();
  }
  if (t == 0) ws[OFF_HSUM] = sh[0];
}

// ---------------- K5: expand 3->56 and build hsmall[b][p][ij(64)][o(56)] ----------------
__global__ __launch_bounds__(256) void k_expand(float* ws, const float* __restrict__ we,
                                                const float* __restrict__ be) {
  int t = blockIdx.x * 256 + threadIdx.x;    // 0..229375
  if (t >= 16 * 4 * 64 * 56) return;
  int o = t % 56;
  int r = t / 56;
  int ij = r % 64;
  int bp = r / 64;                           // b*4 + p == b4
  const float* meas = ws + OFF_MEAS;
  float lvl0 = (float)((ij >> 3) + 1) * meas[bp * 18 + 2 + 0];   // measure[b4][1][0]
  float lvl1 = (float)((ij & 7) + 1) * meas[bp * 18 + 2 + 1];    // measure[b4][1][1]
  float hs = ws[OFF_HSUM];
  float hn = ws[OFF_HIST + bp * 64 + ij] / hs;
  float v = we[o * 3 + 0] * lvl0 + we[o * 3 + 1] * lvl1 + we[o * 3 + 2] * hn + be[o];
  ws[OFF_HSMALL + t] = leaky01(v);
}

// ---------------- K6: LayerNorm stats per (b,p) using replication weights ----------------
__global__ __launch_bounds__(256) void k_stats(float* ws) {
  int bp = blockIdx.x;                       // 0..63
  const float* hs = ws + OFF_HSMALL + (size_t)bp * NSM;
  float s1 = 0.f, s2 = 0.f;
  for (int e = threadIdx.x; e < NSM; e += 256) {
    int o = e % 56;
    int lo = (o * 32 + 6) / 7;
    int hi = ((o + 1) * 32 + 6) / 7;
    float wgt = (float)((hi - lo) * 4);      // replication count in 256x256 image
    float v = hs[e];
    s1 += wgt * v;
    s2 += wgt * v * v;
  }
  __shared__ float a1[256], a2[256];
  int t = threadIdx.x;
  a1[t] = s1; a2[t] = s2;
  __syncthreads();
  for (int k = 128; k > 0; k >>= 1) {
    if (t < k) { a1[t] += a1[t + k]; a2[t] += a2[t + k]; }
    __syncthreads();
  }
  if (t == 0) {
    float mean = a1[0] * (1.0f / 65536.0f);
    float var = fmaxf(a2[0] * (1.0f / 65536.0f) - mean * mean, 0.f);
    ws[OFF_STATS + bp * 2 + 0] = mean;
    ws[OFF_STATS + bp * 2 + 1] = rsqrtf(var + 1e-5f);
  }
}

// ---------------- K7: WMMA channel projection 4 -> 16, fused LayerNorm + bias + leaky ----
// M = 57344 positions, K = 4 (pair dim), N = 16 channels: V_WMMA_F32_16X16X4_F32 per wave.
__global__ __launch_bounds__(256) void k_chan16(float* ws, const float* __restrict__ wconv,
                                                const float* __restrict__ bconv) {
  const float* hsmall = ws + OFF_HSMALL;
  const float* stats = ws + OFF_STATS;
  float* h2 = ws + OFF_H2;
  const int wave = (blockIdx.x * 256 + threadIdx.x) >> 5;  // tile 0..3583 (grid exact)
  const int lane = threadIdx.x & 31;
  const int m = lane & 15;
  const int k0 = (lane >> 4) * 2;            // K index base for this lane half
  const int b = wave / 224;                  // 224 tiles per batch; tiles never cross b
  const int posInB = (wave % 224) * 16 + m;  // position within batch for A row m
  const int bp0 = b * 4 + k0;
  const float mean0 = stats[bp0 * 2 + 0], is0 = stats[bp0 * 2 + 1];
  const float mean1 = stats[bp0 * 2 + 2], is1 = stats[bp0 * 2 + 3];
  v2f a, bm;
  a.x = (hsmall[(size_t)bp0 * NSM + posInB] - mean0) * is0;         // A[m][k0]
  a.y = (hsmall[(size_t)(bp0 + 1) * NSM + posInB] - mean1) * is1;   // A[m][k0+1]
  bm.x = wconv[m * 4 + k0];                  // B[k0][n=m]  (w_conv is (16,4) row-major)
  bm.y = wconv[m * 4 + k0 + 1];              // B[k0+1][n=m]
  v8f c = {0.f, 0.f, 0.f, 0.f, 0.f, 0.f, 0.f, 0.f};
  c = __builtin_amdgcn_wmma_f32_16x16x4_f32(false, a, false, bm, (short)0, c,
                                            false, false);
  const int n = m;                           // C/D: lane%16 == N
  const float bc = bconv[n];
  const int mbase = (lane >> 4) * 8;         // C/D: VGPR r -> M = r (+8 for hi lanes)
  const int tileBase = wave * 16;
  #pragma unroll
  for (int r = 0; r < 8; ++r) {
    float v = leaky01(c[r] + bc);
    h2[(size_t)(tileBase + mbase + r) * 16 + n] = v;
  }
}

// ---------------- K8: fused nearest-upsample + 3x3 conv(16->1) + leaky ----------------
__global__ __launch_bounds__(256) void k_final(const float* __restrict__ ws,
                                               const float* __restrict__ wf,
                                               const float* __restrict__ bf,
                                               float* __restrict__ out) {
  const float* h2 = ws + OFF_H2;
  int idx = blockIdx.x * 256 + threadIdx.x;  // 0..1048575 (grid exact)
  int b = idx >> 16;
  int y = (idx >> 8) & 255;
  int x = idx & 255;
  float acc = bf[0];
  #pragma unroll
  for (int dy = -1; dy <= 1; ++dy) {
    int yy = y + dy;
    if ((unsigned)yy >= 256u) continue;
    int iy = yy >> 2;                        // 256 -> 64 rows (exact /4)
    #pragma unroll
    for (int dx = -1; dx <= 1; ++dx) {
      int xx = x + dx;
      if ((unsigned)xx >= 256u) continue;
      int jx = (xx * 7) >> 5;                // 256 -> 56 cols (floor(x*56/256))
      const float* v = h2 + (size_t)((b * 64 + iy) * 56 + jx) * 16;
      int tap = (dy + 1) * 3 + (dx + 1);
      #pragma unroll
      for (int o = 0; o < 16; ++o) acc += wf[o * 9 + tap] * v[o];
    }
  }
  out[idx] = leaky01(acc);
}

// ---------------- host launch ----------------
extern "C" void kernel_launch(void* const* d_in, const int* in_sizes, int n_in,
                              void* d_out, int out_size, void* d_ws, size_t ws_size,
                              hipStream_t stream) {
  const float* x        = (const float*)d_in[0];
  const float* clamp_lo = (const float*)d_in[1];
  const float* clamp_hi = (const float*)d_in[2];
  const float* w_expand = (const float*)d_in[3];
  const float* b_expand = (const float*)d_in[4];
  const float* w_conv   = (const float*)d_in[5];
  const float* b_conv   = (const float*)d_in[6];
  const float* w_final  = (const float*)d_in[7];
  const float* b_final  = (const float*)d_in[8];
  float* out = (float*)d_out;
  float* ws = (float*)d_ws;
  (void)in_sizes; (void)n_in; (void)out_size; (void)ws_size;

  k_init<<<1, 256, 0, stream>>>(ws);
  k_S<<<dim3((NPIX + 255) / 256, B_), 256, 0, stream>>>(x, ws);
  k_minmax<<<dim3(32, NPLANE), 256, 0, stream>>>(ws, clamp_lo, clamp_hi);
  k_measure<<<1, 128, 0, stream>>>(ws);
  k_hist<<<dim3(32, NB4), 256, 0, stream>>>(ws, clamp_lo, clamp_hi);
  k_histsum<<<1, 256, 0, stream>>>(ws);
  k_expand<<<(16 * 4 * 64 * 56) / 256, 256, 0, stream>>>(ws, w_expand, b_expand);
  k_stats<<<NB4, 256, 0, stream>>>(ws);
  k_chan16<<<NTILE / 8, 256, 0, stream>>>(ws, w_conv, b_conv);   // 8 waves/block
  k_final<<<(16 * 256 * 256) / 256, 256, 0, stream>>>(ws, w_final, b_final, out);
}